// MNN_GNN_16269336118023
// MI455X (gfx1250) — compile-verified
//
#include <hip/hip_runtime.h>
#include <hip/hip_bf16.h>

#define HD 256
#define NCLS 10

typedef __attribute__((ext_vector_type(2))) float v2f;
typedef __attribute__((ext_vector_type(8))) float v8f;

// ---------------------------------------------------------------------------
// Zero the BN stats scratch (2048 floats)
// ---------------------------------------------------------------------------
__global__ __launch_bounds__(256) void zero_stats_kernel(float* __restrict__ stats) {
    int i = blockIdx.x * 256 + threadIdx.x;
    if (i < 2048) stats[i] = 0.0f;
}

// ---------------------------------------------------------------------------
// agg := x   (float4 copy; agg then receives the edge scatter so h = x + sum)
// ---------------------------------------------------------------------------
__global__ __launch_bounds__(256) void copy_init_kernel(const float4* __restrict__ src,
                                                        float4* __restrict__ dst,
                                                        long long n4) {
    long long i = (long long)blockIdx.x * 256 + threadIdx.x;
    if (i < n4) dst[i] = src[i];
}

// ---------------------------------------------------------------------------
// Scatter-add: agg[dst] += x[src].  4 edges per block, 64 lanes/edge, 4 f32
// per lane (float4 gather + 4 global_atomic_add_f32, L2-resident).
// ---------------------------------------------------------------------------
__global__ __launch_bounds__(256) void scatter_add_kernel(const float* __restrict__ x,
                                                          const long long* __restrict__ ei,
                                                          float* __restrict__ agg,
                                                          long long E) {
    long long e = (long long)blockIdx.x * 4 + (threadIdx.x >> 6);
    if (e >= E) return;
    int t = threadIdx.x & 63;
    long long s = ei[e];
    long long d = ei[E + e];
    float4 v = ((const float4*)(x + s * HD))[t];
    float* out = agg + d * HD + t * 4;
    atomicAdd(out + 0, v.x);
    atomicAdd(out + 1, v.y);
    atomicAdd(out + 2, v.z);
    atomicAdd(out + 3, v.w);
}

// ---------------------------------------------------------------------------
// Column sums + sum-of-squares over N rows of a [N,256] matrix (for BN).
// Thread = column, block-strided over rows, atomic reduce at the end.
// ---------------------------------------------------------------------------
__global__ __launch_bounds__(256) void column_stats_kernel(const float* __restrict__ M,
                                                           float* __restrict__ sum,
                                                           float* __restrict__ sumsq,
                                                           int nrows) {
    int col = threadIdx.x;
    float s = 0.0f, ss = 0.0f;
    for (int r = blockIdx.x; r < nrows; r += gridDim.x) {
        float v = M[(long long)r * HD + col];
        s += v;
        ss += v * v;
    }
    atomicAdd(&sum[col], s);
    atomicAdd(&sumsq[col], ss);
}

// ---------------------------------------------------------------------------
// bn(x) = x*scale + bias  with  scale = gamma*rsqrt(var+eps),
//                              bias  = beta - mu*scale   (biased variance)
// ---------------------------------------------------------------------------
__global__ __launch_bounds__(256) void bn_finalize_kernel(const float* __restrict__ sum,
                                                          const float* __restrict__ sumsq,
                                                          const float* __restrict__ gamma,
                                                          const float* __restrict__ beta,
                                                          float* __restrict__ scale,
                                                          float* __restrict__ bias,
                                                          float inv_n) {
    int c = threadIdx.x;
    float mu = sum[c] * inv_n;
    float var = sumsq[c] * inv_n - mu * mu;
    float sc = rsqrtf(var + 1e-5f) * gamma[c];
    scale[c] = sc;
    bias[c] = beta[c] - mu * sc;
}

// ---------------------------------------------------------------------------
// A-fragment load with fused prologue (one v2f = K pair {k, k+1} of row arow)
//   MODE 1: plain                      MODE 2: relu(bn(.))
//   MODE 3: x + 0.01*lrelu(bn4(.))
// ---------------------------------------------------------------------------
template <int MODE>
__device__ __forceinline__ v2f load_a_frag(const float* __restrict__ A,
                                           const float* __restrict__ scaleA,
                                           const float* __restrict__ biasA,
                                           const float* __restrict__ Xres,
                                           long long arow, int k) {
    v2f a;
    if (MODE == 2) {
        float a0 = A[arow + k] * scaleA[k] + biasA[k];
        float a1 = A[arow + k + 1] * scaleA[k + 1] + biasA[k + 1];
        a.x = a0 > 0.0f ? a0 : 0.0f;
        a.y = a1 > 0.0f ? a1 : 0.0f;
    } else if (MODE == 3) {
        float h0 = A[arow + k] * scaleA[k] + biasA[k];
        float h1 = A[arow + k + 1] * scaleA[k + 1] + biasA[k + 1];
        h0 = h0 >= 0.0f ? h0 : 0.1f * h0;
        h1 = h1 >= 0.0f ? h1 : 0.1f * h1;
        a.x = Xres[arow + k] + 0.01f * h0;
        a.y = Xres[arow + k + 1] + 0.01f * h1;
    } else {
        a.x = A[arow + k];
        a.y = A[arow + k + 1];
    }
    return a;
}

// ---------------------------------------------------------------------------
// f32 WMMA GEMM:  Out[N,NCOL] = prologue(A[N,256]) @ W[256,NCOL] + b, epilogue
// Each wave computes a 32x64 tile: 2 row-subtiles x 4 col-subtiles of
// V_WMMA_F32_16X16X4_F32 over K=256.  B fragments are shared by both
// row-subtiles (8 WMMAs per 2 A-loads + 8 B-loads).
// Block = 8 waves: 2x4 (NCOL=256, 64 rows) or 4x2 (NCOL=128, 128 rows).
// ---------------------------------------------------------------------------
template <int MODE, int NCOL>
__global__ __launch_bounds__(256) void gemm_wmma_kernel(const float* __restrict__ A,
                                                        const float* __restrict__ W,
                                                        const float* __restrict__ bvec,
                                                        const float* __restrict__ scaleA,
                                                        const float* __restrict__ biasA,
                                                        const float* __restrict__ Xres,
                                                        float* __restrict__ Out,
                                                        int nrows) {
    constexpr int WAVES_COL = NCOL / 64;       // 4 (NCOL=256) or 2 (NCOL=128)
    constexpr int WAVES_ROW = 8 / WAVES_COL;   // 2 or 4
    constexpr int ROWS = 32 * WAVES_ROW;       // 64 or 128

    const int wid  = threadIdx.x >> 5;
    const int lane = threadIdx.x & 31;
    const int wrow = wid / WAVES_COL;
    const int wcol = wid % WAVES_COL;
    const int i0   = blockIdx.x * ROWS + wrow * 32;
    if (i0 + 16 > nrows) return;                    // wave-uniform (N % 16 == 0)
    const bool both = (i0 + 32 <= nrows);           // second row-subtile valid?
    const int j0   = wcol * 64;

    const int m    = lane & 15;                // A-row-in-tile / B-col-in-tile
    const int half = lane >> 4;
    const int kb   = half * 2;                 // K sub-offset per lane half
    const long long arow0 = (long long)(i0 + m) * HD;
    const long long arow1 = both ? arow0 + 16LL * HD : arow0;  // alias if invalid

    v8f c00 = {}, c01 = {}, c02 = {}, c03 = {};    // row-subtile 0
    v8f c10 = {}, c11 = {}, c12 = {}, c13 = {};    // row-subtile 1

    for (int k0 = 0; k0 < HD; k0 += 4) {
        const int k = k0 + kb;
        v2f a0 = load_a_frag<MODE>(A, scaleA, biasA, Xres, arow0, k);
        v2f a1 = load_a_frag<MODE>(A, scaleA, biasA, Xres, arow1, k);

        const float* w0 = W + (long long)k * NCOL + j0 + m;  // row k
        const float* w1 = w0 + NCOL;                         // row k+1
        v2f b0, b1, b2, b3;
        b0.x = w0[0];  b0.y = w1[0];
        b1.x = w0[16]; b1.y = w1[16];
        b2.x = w0[32]; b2.y = w1[32];
        b3.x = w0[48]; b3.y = w1[48];

        c00 = __builtin_amdgcn_wmma_f32_16x16x4_f32(false, a0, false, b0, (short)0, c00, false, false);
        c10 = __builtin_amdgcn_wmma_f32_16x16x4_f32(false, a1, false, b0, (short)0, c10, false, false);
        c01 = __builtin_amdgcn_wmma_f32_16x16x4_f32(false, a0, false, b1, (short)0, c01, false, false);
        c11 = __builtin_amdgcn_wmma_f32_16x16x4_f32(false, a1, false, b1, (short)0, c11, false, false);
        c02 = __builtin_amdgcn_wmma_f32_16x16x4_f32(false, a0, false, b2, (short)0, c02, false, false);
        c12 = __builtin_amdgcn_wmma_f32_16x16x4_f32(false, a1, false, b2, (short)0, c12, false, false);
        c03 = __builtin_amdgcn_wmma_f32_16x16x4_f32(false, a0, false, b3, (short)0, c03, false, false);
        c13 = __builtin_amdgcn_wmma_f32_16x16x4_f32(false, a1, false, b3, (short)0, c13, false, false);
    }

    // Epilogue + store: C VGPR r holds row (r + 8*half), col j0 + 16*t + m.
#pragma unroll
    for (int s = 0; s < 2; ++s) {
        if (s == 1 && !both) break;
        const v8f* cs0 = s == 0 ? &c00 : &c10;
        const v8f* cs1 = s == 0 ? &c01 : &c11;
        const v8f* cs2 = s == 0 ? &c02 : &c12;
        const v8f* cs3 = s == 0 ? &c03 : &c13;
#pragma unroll
        for (int r = 0; r < 8; ++r) {
            const int orow = i0 + s * 16 + r + 8 * half;
            float* op = Out + (long long)orow * NCOL;
            float vr[4] = {(*cs0)[r], (*cs1)[r], (*cs2)[r], (*cs3)[r]};
#pragma unroll
            for (int t = 0; t < 4; ++t) {
                const int col = j0 + 16 * t + m;
                float v = vr[t] + bvec[col];
                if (MODE == 2)      v = v >= 0.0f ? v : 0.01f * v;  // lrelu(lrelu(v))
                else if (MODE == 3) v = v >= 0.0f ? v : 0.1f  * v;  // lrelu
                op[col] = v;
            }
        }
    }
}

// ---------------------------------------------------------------------------
// Head: out[N,10] = Z[N,128] @ Wl3[128,10] + bl3  (tiny; Wl3 staged in LDS)
// ---------------------------------------------------------------------------
__global__ __launch_bounds__(256) void head_kernel(const float* __restrict__ Z,
                                                   const float* __restrict__ Wl3,
                                                   const float* __restrict__ bl3,
                                                   float* __restrict__ out,
                                                   int nrows) {
    __shared__ float w[128 * NCLS + NCLS];
    for (int i = threadIdx.x; i < 128 * NCLS; i += 256) w[i] = Wl3[i];
    if (threadIdx.x < NCLS) w[128 * NCLS + threadIdx.x] = bl3[threadIdx.x];
    __syncthreads();

    int row = blockIdx.x * 256 + threadIdx.x;
    if (row >= nrows) return;
    float acc[NCLS];
#pragma unroll
    for (int c = 0; c < NCLS; ++c) acc[c] = w[128 * NCLS + c];
    const float* z = Z + (long long)row * 128;
    for (int k = 0; k < 128; ++k) {
        float zv = z[k];
#pragma unroll
        for (int c = 0; c < NCLS; ++c) acc[c] += zv * w[k * NCLS + c];
    }
    float* o = out + (long long)row * NCLS;
#pragma unroll
    for (int c = 0; c < NCLS; ++c) o[c] = acc[c];
}

// ---------------------------------------------------------------------------
extern "C" void kernel_launch(void* const* d_in, const int* in_sizes, int n_in,
                              void* d_out, int out_size, void* d_ws, size_t ws_size,
                              hipStream_t stream) {
    const float*     x      = (const float*)d_in[0];
    const long long* ei     = (const long long*)d_in[1];   // int64 [2,E]
    const float*     W1     = (const float*)d_in[2];
    const float*     b1     = (const float*)d_in[3];
    const float*     g_mlp  = (const float*)d_in[4];
    const float*     be_mlp = (const float*)d_in[5];
    const float*     W2     = (const float*)d_in[6];
    const float*     b2     = (const float*)d_in[7];
    const float*     g4     = (const float*)d_in[8];
    const float*     be4    = (const float*)d_in[9];
    const float*     Wl1    = (const float*)d_in[10];
    const float*     bl1    = (const float*)d_in[11];
    const float*     Wl3    = (const float*)d_in[12];
    const float*     bl3    = (const float*)d_in[13];
    float*           out    = (float*)d_out;

    const int       N = in_sizes[0] / HD;       // 100000
    const long long E = (long long)in_sizes[1] / 2;

    float* B0    = (float*)d_ws;                      // [N,256]: agg/h, later h2
    float* B1    = B0 + (long long)N * HD;            // [N,256]: C1, later Z[N,128]
    float* stats = B1 + (long long)N * HD;            // 2048 floats
    float* sum1 = stats, *sq1 = stats + 256, *sc1 = stats + 512, *bi1 = stats + 768;
    float* sum4 = stats + 1024, *sq4 = stats + 1280, *sc4 = stats + 1536, *bi4 = stats + 1792;
    const float inv_n = 1.0f / (float)N;

    // 0) zero BN stats accumulators
    zero_stats_kernel<<<8, 256, 0, stream>>>(stats);

    // 1) h = x + segment_sum(x[src], dst)
    long long n4 = (long long)N * HD / 4;
    copy_init_kernel<<<(int)((n4 + 255) / 256), 256, 0, stream>>>((const float4*)x, (float4*)B0, n4);
    scatter_add_kernel<<<(int)((E + 3) / 4), 256, 0, stream>>>(x, ei, B0, E);

    // 2) C1 = h @ W1 + b1
    gemm_wmma_kernel<1, 256><<<(N + 63) / 64, 256, 0, stream>>>(B0, W1, b1, nullptr, nullptr, nullptr, B1, N);

    // 3) BN stats over C1 -> scale1/bias1
    column_stats_kernel<<<2048, 256, 0, stream>>>(B1, sum1, sq1, N);
    bn_finalize_kernel<<<1, 256, 0, stream>>>(sum1, sq1, g_mlp, be_mlp, sc1, bi1, inv_n);

    // 4) h2 = lrelu(lrelu(relu(bn(C1)) @ W2 + b2))  -> B0 (reused)
    gemm_wmma_kernel<2, 256><<<(N + 63) / 64, 256, 0, stream>>>(B1, W2, b2, sc1, bi1, nullptr, B0, N);

    // 5) BN4 stats over h2 -> scale4/bias4
    column_stats_kernel<<<2048, 256, 0, stream>>>(B0, sum4, sq4, N);
    bn_finalize_kernel<<<1, 256, 0, stream>>>(sum4, sq4, g4, be4, sc4, bi4, inv_n);

    // 6) Z = lrelu((x + 0.01*lrelu(bn4(h2))) @ Wl1 + bl1)  -> B1 (reused, [N,128])
    gemm_wmma_kernel<3, 128><<<(N + 127) / 128, 256, 0, stream>>>(B0, Wl1, bl1, sc4, bi4, x, B1, N);

    // 7) out = Z @ Wl3 + bl3
    head_kernel<<<(N + 255) / 256, 256, 0, stream>>>(B1, Wl3, bl3, out, N);
}